// GCNNodeClassifier_68281390072334
// MI455X (gfx1250) — compile-verified
//
#include <hip/hip_runtime.h>
#include <hip/hip_bf16.h>

typedef __attribute__((ext_vector_type(2))) float v2f;
typedef __attribute__((ext_vector_type(8))) float v8f;

#define CFEAT 128

// ---------------------------------------------------------------------------
// Degree count: deg[col[e]] += 1 for every edge (self-loop added in dinv pass)
// ---------------------------------------------------------------------------
__global__ void degree_kernel(const int* __restrict__ col, float* __restrict__ deg, int E) {
    int e = blockIdx.x * blockDim.x + threadIdx.x;
    if (e < E) atomicAdd(&deg[col[e]], 1.0f);
}

__global__ void dinv_kernel(float* __restrict__ deg, int n) {
    int i = blockIdx.x * blockDim.x + threadIdx.x;
    if (i < n) {
        float d = deg[i] + 1.0f;   // +1 self-loop; always > 0
        deg[i] = rsqrtf(d);
    }
}

// ---------------------------------------------------------------------------
// fp32 WMMA GEMM: C[M,N] = A[M,K] @ W[K,N] (+ bias), v_wmma_f32_16x16x4_f32.
// K, N compile-time constants -> K-loop fully unrolled, all global loads use
// immediate offsets from two base pointers (A-row, W-column strip).
// One wave computes one 16x16 tile; block = 128 threads = 4 waves = 4 N-tiles.
// grid = (M/16, N/64) exact, so EXEC is all-1s everywhere (WMMA requirement).
// ISA fp32 fragment layouts (CDNA5 ISA §7.12.2), lane = 16*hi + lo:
//   A (16x4): a.x = A[lo][k0+2*hi],     a.y = A[lo][k0+2*hi+1]
//   B (4x16): b.x = W[k0+2*hi][n0+lo],  b.y = W[k0+2*hi+1][n0+lo]
//   C/D (16x16): acc[v] -> row m0+v+8*hi, col n0+lo
// ---------------------------------------------------------------------------
template <int K, int N, int ADD_BIAS>
__global__ void gemm16_wmma_f32(const float* __restrict__ A,
                                const float* __restrict__ W,
                                const float* __restrict__ bias,
                                float* __restrict__ C, int M) {
    const int lane = threadIdx.x & 31;
    const int wave = threadIdx.x >> 5;
    const int lo = lane & 15;
    const int hi = lane >> 4;

    const int m0 = blockIdx.x * 16;
    const int n0 = (blockIdx.y * 4 + wave) * 16;

    // Base pointers: all K-loop accesses are immediate offsets from these.
    const float* abase = A + (size_t)(m0 + lo) * K + 2 * hi;   // + k0
    const float* wbase = W + (size_t)(2 * hi) * N + n0 + lo;   // + k0*N

    v8f acc = {};
#pragma unroll
    for (int k0 = 0; k0 < K; k0 += 4) {
        v2f a, b;
        a.x = abase[k0];            // contiguous pair -> global_load_b64
        a.y = abase[k0 + 1];
        b.x = wbase[(size_t)k0 * N];
        b.y = wbase[(size_t)(k0 + 1) * N];
        acc = __builtin_amdgcn_wmma_f32_16x16x4_f32(
            /*neg_a=*/false, a, /*neg_b=*/false, b,
            /*c_mod=*/(short)0, acc, /*reuse_a=*/false, /*reuse_b=*/false);
    }

    const float bv = ADD_BIAS ? bias[n0 + lo] : 0.0f;
    float* cbase = C + (size_t)(m0 + 8 * hi) * N + n0 + lo;
#pragma unroll
    for (int v = 0; v < 8; ++v) {
        cbase[(size_t)v * N] = acc[v] + bv;
    }
}

// ---------------------------------------------------------------------------
// Edge aggregation: one wave per edge. 128 features = 32 lanes x float4.
// agg[col[e]] += h[row[e]] * dinv[row]*dinv[col]   via global_atomic_add_f32.
// ---------------------------------------------------------------------------
__global__ void aggregate_edges(const float* __restrict__ h,
                                const int* __restrict__ row,
                                const int* __restrict__ col,
                                const float* __restrict__ dinv,
                                float* __restrict__ agg, int E) {
    const int lane = threadIdx.x & 31;
    const int e = blockIdx.x * (blockDim.x >> 5) + (threadIdx.x >> 5);
    if (e >= E) return;
    const int r = row[e];
    const int c = col[e];
    const float nrm = dinv[r] * dinv[c];
    const float4 v = ((const float4*)(h + (size_t)r * CFEAT))[lane];
    float* dst = agg + (size_t)c * CFEAT + lane * 4;
    atomicAdd(dst + 0, v.x * nrm);
    atomicAdd(dst + 1, v.y * nrm);
    atomicAdd(dst + 2, v.z * nrm);
    atomicAdd(dst + 3, v.w * nrm);
}

// ---------------------------------------------------------------------------
// Self-loop contribution + bias + ReLU (in place on agg):
//   agg[i,f] = relu(agg[i,f] + h[i,f]*dinv[i]^2 + bias[f])
// ---------------------------------------------------------------------------
__global__ void selfloop_bias_relu(const float* __restrict__ h,
                                   const float* __restrict__ dinv,
                                   const float* __restrict__ bias,
                                   float* __restrict__ agg,
                                   int n) {
    const size_t idx = (size_t)blockIdx.x * blockDim.x + threadIdx.x;
    if (idx >= (size_t)n * CFEAT) return;
    const int i = (int)(idx >> 7);        // /128
    const int f = (int)(idx & 127);       // %128
    const float d = dinv[i];
    const float v = agg[idx] + h[idx] * d * d + bias[f];
    agg[idx] = v > 0.0f ? v : 0.0f;
}

// ---------------------------------------------------------------------------
// Host launch
// ---------------------------------------------------------------------------
extern "C" void kernel_launch(void* const* d_in, const int* in_sizes, int n_in,
                              void* d_out, int out_size, void* d_ws, size_t ws_size,
                              hipStream_t stream) {
    const float* x  = (const float*)d_in[0];
    const int*   ei = (const int*)d_in[1];     // [2, E] row-major (int32 from JAX)
    const float* W1 = (const float*)d_in[2];
    const float* b1 = (const float*)d_in[3];
    const float* W2 = (const float*)d_in[4];
    const float* b2 = (const float*)d_in[5];
    const float* Wl = (const float*)d_in[6];
    const float* bl = (const float*)d_in[7];
    float* out = (float*)d_out;

    const int E  = in_sizes[1] / 2;
    const int Nn = in_sizes[0] / CFEAT;        // 40000
    const int C  = CFEAT;                      // 128

    const int* row = ei;                       // edge_index[0] = source
    const int* col = ei + E;                   // edge_index[1] = target

    // Workspace layout: dinv | bufA (N*C) | bufB (N*C)
    float* dinv = (float*)d_ws;
    float* bufA = dinv + Nn;                   // Nn*4 bytes keeps 16B alignment
    float* bufB = bufA + (size_t)Nn * C;

    const size_t featBytes = (size_t)Nn * C * sizeof(float);
    const int Mt = Nn / 16;                    // 2500 row tiles
    const int featBlocks = (int)(((size_t)Nn * C + 255) / 256);

    // --- normalization ---
    hipMemsetAsync(dinv, 0, (size_t)Nn * sizeof(float), stream);
    degree_kernel<<<(E + 255) / 256, 256, 0, stream>>>(col, dinv, E);
    dinv_kernel<<<(Nn + 255) / 256, 256, 0, stream>>>(dinv, Nn);

    // --- layer 1: h1 = x @ W1 ---
    gemm16_wmma_f32<128, 128, 0><<<dim3(Mt, 2), 128, 0, stream>>>(x, W1, nullptr, bufA, Nn);
    hipMemsetAsync(bufB, 0, featBytes, stream);
    aggregate_edges<<<(E + 7) / 8, 256, 0, stream>>>(bufA, row, col, dinv, bufB, E);
    selfloop_bias_relu<<<featBlocks, 256, 0, stream>>>(bufA, dinv, b1, bufB, Nn);

    // --- layer 2: h2 = x1 @ W2 ---
    gemm16_wmma_f32<128, 128, 0><<<dim3(Mt, 2), 128, 0, stream>>>(bufB, W2, nullptr, bufA, Nn);
    hipMemsetAsync(bufB, 0, featBytes, stream);   // x1 no longer needed
    aggregate_edges<<<(E + 7) / 8, 256, 0, stream>>>(bufA, row, col, dinv, bufB, E);
    selfloop_bias_relu<<<featBlocks, 256, 0, stream>>>(bufA, dinv, b2, bufB, Nn);

    // --- classifier: out = x2 @ Wl + bl ---
    gemm16_wmma_f32<128, 64, 1><<<dim3(Mt, 1), 128, 0, stream>>>(bufB, Wl, bl, out, Nn);
}